// UpsampleLoss_63539746177641
// MI455X (gfx1250) — compile-verified
//
#include <hip/hip_runtime.h>
#include <hip/hip_bf16.h>

// CDNA5 / gfx1250 wave32 WMMA f32 16x16x4 types
typedef float v2f_t __attribute__((ext_vector_type(2)));
typedef float v8f_t __attribute__((ext_vector_type(8)));

#define FRAME_X 222
#define FRAME_Y 124
#define GRID_CELLS (FRAME_X * FRAME_Y)

// Workspace layout (floats):
// [0..3]   accF  (sum of row-mins, gt->pred) per batch
// [4..7]   accB  (sum of row-mins, pred->gt) per batch
// [8..15]  (as uint) minmax: pred{minx,miny,maxx,maxy}, gt{minx,miny,maxx,maxy}
// [16..16+27528) diff grid (pred_frame - gt_frame)
#define WS_ACCF 0
#define WS_ACCB 4
#define WS_MM   8
#define WS_GRID 16

// ---------- ordered-uint encoding for float atomic min/max ----------
__device__ __forceinline__ unsigned f2ord(float f) {
    unsigned u = __float_as_uint(f);
    return (u & 0x80000000u) ? ~u : (u | 0x80000000u);
}
__device__ __forceinline__ float ord2f(unsigned e) {
    unsigned u = (e & 0x80000000u) ? (e ^ 0x80000000u) : ~e;
    return __uint_as_float(u);
}

// ---------- init: zero accumulators + grid, neutral-init minmax ----------
__global__ void init_ws_kernel(float* ws) {
    unsigned* u = (unsigned*)ws;
    int total = WS_GRID + GRID_CELLS;
    for (int k = blockIdx.x * blockDim.x + threadIdx.x; k < total;
         k += gridDim.x * blockDim.x) {
        if (k >= WS_MM && k < WS_GRID) {
            int r = (k - WS_MM) & 3;       // 0,1 = min slots ; 2,3 = max slots
            u[k] = (r < 2) ? 0xFFFFFFFFu : 0u;
        } else {
            ws[k] = 0.0f;
        }
    }
}

// ---------- chamfer: one wave per 16-row tile of A; scans all of Bcols ----------
// dist tile via  D = (-2*A) x B + C  on V_WMMA_F32_16X16X4_F32, C holding the
// norm terms.  Every lane loads a FULL point (both halves read the same 16
// column points -> coalesced, L2-resident), so the inner loop has no lane
// divergence and no cross-lane shuffles.
__global__ __launch_bounds__(32)
void chamfer_min_kernel(const float* __restrict__ Arows,  // [B, MA, 3]
                        const float* __restrict__ Bcols,  // [B, NB, 3]
                        float* __restrict__ acc,          // [B]
                        int MA, int NB) {
    const int mtiles = MA >> 4;
    const int b  = blockIdx.x / mtiles;
    const int mt = blockIdx.x % mtiles;
    const int lane = threadIdx.x;        // 0..31 (wave32)
    const int half = lane >> 4;          // 0: K=0,1 ; 1: K=2,3
    const int l16  = lane & 15;

    // Load this wave's A row l16 fully in every lane (halves duplicate).
    const float* ap = Arows + ((size_t)b * MA + (size_t)(mt * 16 + l16)) * 3;
    const float ax = ap[0], ay = ap[1], az = ap[2];
    const float na = ax * ax + ay * ay + az * az;   // ||A_row[l16]||^2, per lane

    // A operand (16x4 f32, 2 VGPRs): lanes 0-15 K={0,1}=x,y ; lanes 16-31 K={2,3}=z,0
    v2f_t A;
    A.x = -2.0f * (half ? az : ax);
    A.y = half ? 0.0f : (-2.0f * ay);

    // Per-lane selected row norms for the C matrix: M = v + 8*half
    float ng[8];
#pragma unroll
    for (int v = 0; v < 8; ++v) {
        float lo = __shfl(na, v, 32);        // ||A_row[v]||^2    (readlane, uniform)
        float hi = __shfl(na, v + 8, 32);    // ||A_row[v+8]||^2
        ng[v] = half ? hi : lo;
    }

    v8f_t minacc;
#pragma unroll
    for (int v = 0; v < 8; ++v) minacc[v] = 3.0e38f;

    const float* bp = Bcols + ((size_t)b * NB + (size_t)l16) * 3;
    for (int nt = 0; nt < NB; nt += 16) {
        const float bx = bp[0], by = bp[1], bz = bp[2];
        __builtin_prefetch(bp + 96, 0, 1);          // 2 tiles ahead -> global_prefetch

        const float nb = bx * bx + by * by + bz * bz;   // ||B_col[l16]||^2, per lane

        // B operand (4x16 f32, 2 VGPRs): lanes 0-15 rows K={0,1}; lanes 16-31 K={2,3}
        v2f_t Bm;
        Bm.x = half ? bz : bx;
        Bm.y = half ? 0.0f : by;

        v8f_t C;
#pragma unroll
        for (int v = 0; v < 8; ++v) C[v] = ng[v] + nb;

        // D[m][n] = ||a_m||^2 + ||b_n||^2 - 2 a_m . b_n
        v8f_t D = __builtin_amdgcn_wmma_f32_16x16x4_f32(
            false, A, false, Bm, (short)0, C, false, false);

#pragma unroll
        for (int v = 0; v < 8; ++v) minacc[v] = fminf(minacc[v], D[v]);

        bp += 48;   // 16 points * 3 floats
    }

    // Row-min reduction: min across the 16 lanes of each half (per VGPR v),
    // then sum the 16 row-mins.
    float s = 0.0f;
#pragma unroll
    for (int v = 0; v < 8; ++v) {
        float m = minacc[v];
        m = fminf(m, __shfl_xor(m, 1, 32));
        m = fminf(m, __shfl_xor(m, 2, 32));
        m = fminf(m, __shfl_xor(m, 4, 32));
        m = fminf(m, __shfl_xor(m, 8, 32));
        s += m;                 // every lane of a half now holds that half's row-min
    }
    s += __shfl_xor(s, 16, 32); // sum of all 16 row-mins of this tile
    if (lane == 0) atomicAdd(&acc[b], s);
}

// ---------- global min/max of channels 1:3 (ordered-uint atomics) ----------
__global__ __launch_bounds__(256)
void minmax_kernel(const float* __restrict__ pts, int npts, unsigned* mm) {
    int i = blockIdx.x * blockDim.x + threadIdx.x;
    unsigned enx = 0xFFFFFFFFu, eny = 0xFFFFFFFFu, exx = 0u, exy = 0u;
    if (i < npts) {
        float x = pts[(size_t)i * 3 + 1];
        float y = pts[(size_t)i * 3 + 2];
        unsigned ex = f2ord(x), ey = f2ord(y);
        enx = ex; eny = ey; exx = ex; exy = ey;
    }
#pragma unroll
    for (int m = 1; m < 32; m <<= 1) {
        enx = min(enx, (unsigned)__shfl_xor((int)enx, m, 32));
        eny = min(eny, (unsigned)__shfl_xor((int)eny, m, 32));
        exx = max(exx, (unsigned)__shfl_xor((int)exx, m, 32));
        exy = max(exy, (unsigned)__shfl_xor((int)exy, m, 32));
    }
    if ((threadIdx.x & 31) == 0) {
        atomicMin(&mm[0], enx);
        atomicMin(&mm[1], eny);
        atomicMax(&mm[2], exx);
        atomicMax(&mm[3], exy);
    }
}

// ---------- scatter ±1 into the diff grid ----------
__global__ __launch_bounds__(256)
void scatter_kernel(const float* __restrict__ pts, int npts,
                    const unsigned* __restrict__ mm, float* __restrict__ grid,
                    float sign) {
    int i = blockIdx.x * blockDim.x + threadIdx.x;
    if (i >= npts) return;
    float x = pts[(size_t)i * 3 + 1];
    float y = pts[(size_t)i * 3 + 2];
    float minx = ord2f(mm[0]), miny = ord2f(mm[1]);
    float maxx = ord2f(mm[2]), maxy = ord2f(mm[3]);
    float ix = rintf((x - minx) / (maxx - minx) * (float)(FRAME_X - 1));
    float iy = rintf((y - miny) / (maxy - miny) * (float)(FRAME_Y - 1));
    int gx = min(max((int)ix, 0), FRAME_X - 1);
    int gy = min(max((int)iy, 0), FRAME_Y - 1);
    atomicAdd(&grid[gx * FRAME_Y + gy], sign);
}

// ---------- finalize: grid MSE + chamfer combine ----------
__global__ __launch_bounds__(256)
void final_kernel(const float* __restrict__ accF, const float* __restrict__ accB,
                  const float* __restrict__ radius, const float* __restrict__ grid,
                  float* __restrict__ out, int B, int MA, int NB) {
    __shared__ float sh[256];
    float s = 0.0f;
    for (int i = threadIdx.x; i < GRID_CELLS; i += 256) {
        float g = grid[i];
        s += g * g;
    }
    sh[threadIdx.x] = s;
    __syncthreads();
    for (int st = 128; st > 0; st >>= 1) {
        if (threadIdx.x < st) sh[threadIdx.x] += sh[threadIdx.x + st];
        __syncthreads();
    }
    if (threadIdx.x == 0) {
        float cd = 0.0f;
        for (int b = 0; b < B; ++b) {
            float cf = accF[b] / (float)MA;   // mean over gt points
            float cb = accB[b] / (float)NB;   // mean over pred points
            cd += (0.8f * cf + 0.2f * cb) / radius[b];
        }
        cd = (cd / (float)B) * 100.0f;
        float fl = sh[0] / (float)GRID_CELLS; // BETA = 1
        out[0] = cd + fl;
    }
}

extern "C" void kernel_launch(void* const* d_in, const int* in_sizes, int n_in,
                              void* d_out, int out_size, void* d_ws, size_t ws_size,
                              hipStream_t stream) {
    const float* pred = (const float*)d_in[0];  // [B, NP, 3]
    const float* gt   = (const float*)d_in[1];  // [B, NG, 3]
    const float* rad  = (const float*)d_in[2];  // [B]
    float* out = (float*)d_out;

    const int B   = in_sizes[2];                // 4
    const int NPt = in_sizes[0] / 3;            // total pred points (32768)
    const int NGt = in_sizes[1] / 3;            // total gt points   (8192)
    const int NP  = NPt / B;                    // 8192 per batch
    const int NG  = NGt / B;                    // 2048 per batch

    float* ws   = (float*)d_ws;
    float* accF = ws + WS_ACCF;
    float* accB = ws + WS_ACCB;
    unsigned* mm = (unsigned*)(ws + WS_MM);     // pred mm = mm[0..3], gt mm = mm[4..7]
    float* grid = ws + WS_GRID;

    init_ws_kernel<<<64, 256, 0, stream>>>(ws);

    // cost_for: rows = gt (min over pred columns)
    chamfer_min_kernel<<<B * (NG / 16), 32, 0, stream>>>(gt, pred, accF, NG, NP);
    // cost_bac: rows = pred (min over gt columns)
    chamfer_min_kernel<<<B * (NP / 16), 32, 0, stream>>>(pred, gt, accB, NP, NG);

    minmax_kernel<<<(NPt + 255) / 256, 256, 0, stream>>>(pred, NPt, mm);
    minmax_kernel<<<(NGt + 255) / 256, 256, 0, stream>>>(gt, NGt, mm + 4);

    scatter_kernel<<<(NPt + 255) / 256, 256, 0, stream>>>(pred, NPt, mm, grid, 1.0f);
    scatter_kernel<<<(NGt + 255) / 256, 256, 0, stream>>>(gt, NGt, mm + 4, grid, -1.0f);

    final_kernel<<<1, 256, 0, stream>>>(accF, accB, rad, grid, out, B, NG, NP);
}